// LinearSelfAttention_57174604644757
// MI455X (gfx1250) — compile-verified
//
#include <hip/hip_runtime.h>

typedef float v2f __attribute__((ext_vector_type(2)));
typedef float v8f __attribute__((ext_vector_type(8)));

#define S_LEN 2048
#define HEADS 16
#define DHEAD 64
#define SROW  (3 * HEADS * DHEAD)   // 3072 floats between consecutive tokens
#define OROW  (HEADS * DHEAD)       // 1024 floats between consecutive output tokens
#define BQ    128                   // query rows per workgroup (8 waves x 16)
#define BK    64                    // key rows staged per iteration
#define KPITCH 68                   // K tile row pitch: 272B = 17*16B (b128-aligned, bank-skew 4)
#define VPITCH 66                   // Vt tile row pitch: 264B (b64-aligned, bank-skew 2)

__device__ __forceinline__ unsigned lds_off(const void* p) {
    // generic LDS pointer: low 32 bits are the wave-relative LDS byte offset
    return (unsigned)(unsigned long long)p;
}
__device__ __forceinline__ void async_b128(unsigned lds, const float* g) {
    asm volatile("global_load_async_to_lds_b128 %0, %1, off"
                 :: "v"(lds), "v"(g) : "memory");
}
__device__ __forceinline__ void async_b32(unsigned lds, const float* g) {
    asm volatile("global_load_async_to_lds_b32 %0, %1, off"
                 :: "v"(lds), "v"(g) : "memory");
}
__device__ __forceinline__ void wait_async0() {
    asm volatile("s_wait_asynccnt 0" ::: "memory");
}

__launch_bounds__(256)
__global__ void lsa_wmma_f32_kernel(const float* __restrict__ qkv,
                                    float* __restrict__ out) {
    const int qblk = blockIdx.x;        // 0..15
    const int h    = blockIdx.y;        // 0..15
    const int b    = blockIdx.z;
    const int tid  = threadIdx.x;
    // wave id is uniform across the wave: pin it to an SGPR so all dependent
    // control flow (j-loop bound, diagonal test) compiles to scalar branches,
    // not exec-mask divergence (which forced 32 VGPR accumulator copies/iter).
    const int wave = __builtin_amdgcn_readfirstlane(tid >> 5);   // 0..7
    const int lane = tid & 31;
    const int lrow = lane & 15;
    const int lhi  = lane >> 4;

    __shared__ float ldsK0[BK * KPITCH];
    __shared__ float ldsK1[BK * KPITCH];
    __shared__ float ldsVt0[DHEAD * VPITCH];   // V transposed: Vt[d][key]
    __shared__ float ldsVt1[DHEAD * VPITCH];
    __shared__ float ldsS[8 * 16 * 18];        // per-wave S-transpose scratch

    const size_t bbase = (size_t)b * S_LEN * SROW;
    const float* Qb = qkv + bbase + 0 * OROW + h * DHEAD;
    const float* Kb = qkv + bbase + 1 * OROW + h * DHEAD;
    const float* Vb = qkv + bbase + 2 * OROW + h * DHEAD;

    const int qbase = qblk * BQ + wave * 16;   // uniform (SGPR) per wave
    const int qrow  = qbase + lrow;

    // staging index maps (per thread)
    const int ksr = (tid >> 4);         // +16 per 'it' step : K row for b128 copies
    const int ksc = (tid & 15) << 2;    // K col (float4)
    const int vsr = tid & 63;           // V row for transposed b32 copies
    const int vsc = (tid >> 6) << 4;    // first of 16 V cols

    // ---- Q tile (16 x 64) in registers, WMMA A-layout ----
    v2f Aq[16];
    {
        const float* qp = Qb + (size_t)qrow * SROW + 2 * lhi;
#pragma unroll
        for (int c = 0; c < 16; ++c)
            Aq[c] = *(const v2f*)(qp + c * 4);
    }

    v8f acc[4] = {};
    float* sw = &ldsS[wave * (16 * 18)];

    const int kbmax = 2 * qblk + 1;

    // ---- prologue: async-stage block 0 into buffer 0 ----
    {
#pragma unroll
        for (int it = 0; it < 4; ++it) {
            const int r = ksr + it * 16;
            async_b128(lds_off(&ldsK0[r * KPITCH + ksc]),
                       Kb + (size_t)r * SROW + ksc);
        }
#pragma unroll
        for (int t = 0; t < 16; ++t) {
            const int c = vsc + t;
            async_b32(lds_off(&ldsVt0[c * VPITCH + vsr]),
                      Vb + (size_t)vsr * SROW + c);
        }
    }

    for (int kb = 0; kb <= kbmax; ++kb) {
        wait_async0();        // my async group targeting buffer (kb&1) is done
        __syncthreads();      // everyone's group done; prior readers of other buffer done

        const float* kbuf = (kb & 1) ? ldsK1  : ldsK0;
        const float* vbuf = (kb & 1) ? ldsVt1 : ldsVt0;

        // ---- issue async stage of block kb+1 into the other buffer ----
        if (kb < kbmax) {
            float* nk = (kb & 1) ? ldsK0  : ldsK1;
            float* nv = (kb & 1) ? ldsVt0 : ldsVt1;
            const int krow0 = (kb + 1) * BK;
#pragma unroll
            for (int it = 0; it < 4; ++it) {
                const int r = ksr + it * 16;
                async_b128(lds_off(&nk[r * KPITCH + ksc]),
                           Kb + (size_t)(krow0 + r) * SROW + ksc);
            }
#pragma unroll
            for (int t = 0; t < 16; ++t) {
                const int c = vsc + t;
                async_b32(lds_off(&nv[c * VPITCH + vsr]),
                          Vb + (size_t)(krow0 + vsr) * SROW + c);
            }
        }

        // ---- compute on current buffer; trip count is scalar per wave ----
        // sub-blocks j with kb*64 + j*16 <= qbase
        int jcnt = (qbase - kb * BK) / 16 + 1;      // SGPR arithmetic
        if (jcnt > 4) jcnt = 4;
        const bool diagHere = (qbase >= kb * BK) && (qbase < (kb + 1) * BK);

        for (int j = 0; j < jcnt; ++j) {
            // S = Q * K^T (16x16), D=64 as 16 chunks of K=4
            v8f s = {};
#pragma unroll
            for (int c = 0; c < 16; ++c) {
                v2f bk = *(const v2f*)&kbuf[(j * 16 + lrow) * KPITCH + c * 4 + 2 * lhi];
                s = __builtin_amdgcn_wmma_f32_16x16x4_f32(
                        false, Aq[c], false, bk, (short)0, s, false, false);
            }

            // causal mask on the diagonal block (scalar branch; per-lane cndmask)
            if (diagHere && j == jcnt - 1) {
#pragma unroll
                for (int r = 0; r < 8; ++r)
                    if (lrow > r + 8 * lhi) s[r] = 0.0f;
            }

            // transpose S: C/D layout -> A layout via per-wave LDS scratch
#pragma unroll
            for (int r = 0; r < 8; ++r)
                sw[(r + 8 * lhi) * 18 + lrow] = s[r];
            v2f As[4];
#pragma unroll
            for (int c2 = 0; c2 < 4; ++c2)
                As[c2] = *(const v2f*)&sw[lrow * 18 + c2 * 4 + 2 * lhi];

            // O += S * V : B operand is a contiguous b64 read from transposed Vt
#pragma unroll
            for (int dt = 0; dt < 4; ++dt) {
#pragma unroll
                for (int c2 = 0; c2 < 4; ++c2) {
                    v2f bv = *(const v2f*)&vbuf[(dt * 16 + lrow) * VPITCH +
                                                j * 16 + c2 * 4 + 2 * lhi];
                    acc[dt] = __builtin_amdgcn_wmma_f32_16x16x4_f32(
                                  false, As[c2], false, bv, (short)0, acc[dt], false, false);
                }
            }
        }
    }

    // ---- write O tile: element (m = r + 8*lhi, d = dt*16 + lrow) ----
    float* op = out + ((size_t)b * S_LEN + qbase) * OROW + h * DHEAD;
#pragma unroll
    for (int dt = 0; dt < 4; ++dt) {
#pragma unroll
        for (int r = 0; r < 8; ++r)
            op[(size_t)(r + 8 * lhi) * OROW + dt * 16 + lrow] = acc[dt][r];
    }
}

extern "C" void kernel_launch(void* const* d_in, const int* in_sizes, int n_in,
                              void* d_out, int out_size, void* d_ws, size_t ws_size,
                              hipStream_t stream) {
    const float* qkv = (const float*)d_in[0];
    float* out = (float*)d_out;
    const int B = in_sizes[0] / (S_LEN * 3 * HEADS * DHEAD);
    dim3 grid(S_LEN / BQ, HEADS, B);
    lsa_wmma_f32_kernel<<<grid, dim3(256), 0, stream>>>(qkv, out);
}